// PatchTrainer_21775484190923
// MI455X (gfx1250) — compile-verified
//
#include <hip/hip_runtime.h>

// Problem constants (match reference)
#define S      4096
#define NDOTS  100
#define PXPT   4                 // pixels per thread (float4)
#define TPB    256               // 8 waves (wave32)
#define PXPB   (PXPT * TPB)      // 1024 pixels per block

typedef float v4f __attribute__((ext_vector_type(4)));

__global__ __launch_bounds__(TPB) void dots_fill_kernel(
    const float* __restrict__ img,      // [3][S][S]
    const float* __restrict__ centers,  // [NDOTS][2]  (x, y)
    const float* __restrict__ radii,    // [NDOTS]
    const float* __restrict__ colors,   // [NDOTS][3]
    float* __restrict__ out)            // [3][S][S]
{
    __shared__ float sXC[NDOTS];   // compacted dot center x (float, integral)
    __shared__ float sRX2[NDOTS];  // compacted r^2 - dy^2 for this row
    __shared__ float sCR[NDOTS], sCG[NDOTS], sCB[NDOTS];
    __shared__ int   sN;

    const int   y   = blockIdx.y;
    const float fy  = (float)y;
    const int   tid = threadIdx.x;

    // ---- Per-row active-dot compaction (wave 0 only; order-preserving) ----
    if (tid < 32) {
        int base = 0;
        for (int g = 0; g < NDOTS; g += 32) {
            const int i = g + tid;
            bool  act = false;
            float xc  = 0.0f, rx2 = 0.0f;
            if (i < NDOTS) {
                // floor() on non-negative == int() truncation in the reference
                xc          = floorf(centers[2 * i]     * (float)S);
                float yc    = floorf(centers[2 * i + 1] * (float)S);
                float r     = floorf(radii[i] * ((float)S / 5.0f));
                float dy    = fy - yc;
                // exact in fp32: r^2 <= 819^2, dy^2 <= 4095^2 < 2^24, diff < 2^24
                rx2 = r * r - dy * dy;
                act = (rx2 >= 0.0f);   // dot's circle intersects this row
            }
            unsigned m   = (unsigned)__ballot(act);           // wave32 mask
            int      pre = __popc(m & ((1u << tid) - 1u));
            if (act) {
                int p   = base + pre;
                sXC[p]  = xc;
                sRX2[p] = rx2;
                sCR[p]  = colors[3 * i];
                sCG[p]  = colors[3 * i + 1];
                sCB[p]  = colors[3 * i + 2];
            }
            base += __popc(m);
        }
        if (tid == 0) sN = base;
    }
    __syncthreads();
    const int nAct = sN;

    // ---- Per-pixel winner scan (last active dot wins) ----
    const int   x0  = blockIdx.x * PXPB + tid * PXPT;
    const float fx0 = (float)x0;

    int w0 = -1, w1 = -1, w2 = -1, w3 = -1;
    for (int j = 0; j < nAct; ++j) {
        const float xc  = sXC[j];
        const float rx2 = sRX2[j];
        const float d0  = fx0 - xc;        // exact integer-valued fp32
        const float d1  = d0 + 1.0f;
        const float d2  = d0 + 2.0f;
        const float d3  = d0 + 3.0f;
        if (d0 * d0 <= rx2) w0 = j;        // v_cndmask per pixel
        if (d1 * d1 <= rx2) w1 = j;
        if (d2 * d2 <= rx2) w2 = j;
        if (d3 * d3 <= rx2) w3 = j;
    }

    // ---- Load input only where some pixel is uncovered (exec-masked, NT) ----
    const size_t p     = (size_t)y * S + (size_t)x0;
    const size_t plane = (size_t)S * (size_t)S;

    v4f i0 = {0.f, 0.f, 0.f, 0.f}, i1 = i0, i2 = i0;
    const bool need = (w0 < 0) | (w1 < 0) | (w2 < 0) | (w3 < 0);
    if (need) {
        i0 = __builtin_nontemporal_load((const v4f*)(img + p));
        i1 = __builtin_nontemporal_load((const v4f*)(img + plane + p));
        i2 = __builtin_nontemporal_load((const v4f*)(img + 2 * plane + p));
    }

    // ---- Compose: gather winner color from LDS, else passthrough ----
    v4f o0, o1, o2;
    {
        const int  k   = (w0 < 0) ? 0 : w0;
        const bool hit = (w0 >= 0);
        o0.x = hit ? sCR[k] : i0.x;
        o1.x = hit ? sCG[k] : i1.x;
        o2.x = hit ? sCB[k] : i2.x;
    }
    {
        const int  k   = (w1 < 0) ? 0 : w1;
        const bool hit = (w1 >= 0);
        o0.y = hit ? sCR[k] : i0.y;
        o1.y = hit ? sCG[k] : i1.y;
        o2.y = hit ? sCB[k] : i2.y;
    }
    {
        const int  k   = (w2 < 0) ? 0 : w2;
        const bool hit = (w2 >= 0);
        o0.z = hit ? sCR[k] : i0.z;
        o1.z = hit ? sCG[k] : i1.z;
        o2.z = hit ? sCB[k] : i2.z;
    }
    {
        const int  k   = (w3 < 0) ? 0 : w3;
        const bool hit = (w3 >= 0);
        o0.w = hit ? sCR[k] : i0.w;
        o1.w = hit ? sCG[k] : i1.w;
        o2.w = hit ? sCB[k] : i2.w;
    }

    // ---- Streaming NT stores (write-once, > L2 capacity) ----
    __builtin_nontemporal_store(o0, (v4f*)(out + p));
    __builtin_nontemporal_store(o1, (v4f*)(out + plane + p));
    __builtin_nontemporal_store(o2, (v4f*)(out + 2 * plane + p));
}

extern "C" void kernel_launch(void* const* d_in, const int* in_sizes, int n_in,
                              void* d_out, int out_size, void* d_ws, size_t ws_size,
                              hipStream_t stream) {
    const float* img     = (const float*)d_in[0];  // adv_patch [3,4096,4096]
    const float* centers = (const float*)d_in[1];  // [100,2]
    const float* radii   = (const float*)d_in[2];  // [100]
    const float* colors  = (const float*)d_in[3];  // [100,3]
    float*       out     = (float*)d_out;          // [3,4096,4096]

    dim3 grid(S / PXPB, S, 1);   // (4, 4096)
    dim3 block(TPB, 1, 1);
    dots_fill_kernel<<<grid, block, 0, stream>>>(img, centers, radii, colors, out);
}